// RevMaskedLinear_558345749221
// MI455X (gfx1250) — compile-verified
//
#include <hip/hip_runtime.h>
#include <math.h>

// Problem shape (matches reference setup_inputs)
#define BATCH 8192
#define IN_F  2048
#define OUT_F 2048

typedef float v2f __attribute__((ext_vector_type(2)));
typedef float v8f __attribute__((ext_vector_type(8)));

// Exact parameter types for __builtin_amdgcn_global_load_async_to_lds_b128:
//   (v4i __device__* /*AS1 global src*/, v4i AS3* /*LDS dst*/, int off, int cpol)
typedef int v4i_raw __attribute__((vector_size(4 * sizeof(int))));
typedef __attribute__((address_space(1))) v4i_raw gv4i_t;
typedef __attribute__((address_space(3))) v4i_raw lv4i_t;
#define ASG(p) ((gv4i_t*)(void*)(p))
#define ASL(p) ((lv4i_t*)(void*)(p))

// ---------------------------------------------------------------------------
// Phase 1: Bt[i, o] = rev_w[o,i] * softmax_i(mask_w[o,:]/1e-6) * weight[i,o]
// Bt is (IN_F x OUT_F) row-major (K-major, N contiguous) for the GEMM B operand.
// ---------------------------------------------------------------------------
__global__ __launch_bounds__(256) void revmask_prep(
    const float* __restrict__ weight,   // (IN_F, OUT_F) row-major
    const float* __restrict__ rev_w,    // (OUT_F, IN_F) row-major
    const float* __restrict__ mask_w,   // (OUT_F, IN_F) row-major
    float* __restrict__ Bt)             // (IN_F, OUT_F) row-major
{
  __shared__ float red[256];
  const int o = blockIdx.x;
  const int t = threadIdx.x;
  float x[8];
  float m = -INFINITY;
#pragma unroll
  for (int j = 0; j < 8; ++j) {
    x[j] = mask_w[o * IN_F + t + j * 256];
    m = fmaxf(m, x[j]);
  }
  red[t] = m;
  __syncthreads();
  for (int s = 128; s > 0; s >>= 1) {
    if (t < s) red[t] = fmaxf(red[t], red[t + s]);
    __syncthreads();
  }
  const float rowmax = red[0];
  __syncthreads();

  float p[8];
  float sum = 0.0f;
#pragma unroll
  for (int j = 0; j < 8; ++j) {
    p[j] = __expf((x[j] - rowmax) * 1.0e6f);  // /TEMP, TEMP = 1e-6
    sum += p[j];
  }
  red[t] = sum;
  __syncthreads();
  for (int s = 128; s > 0; s >>= 1) {
    if (t < s) red[t] = red[t] + red[t + s];
    __syncthreads();
  }
  const float inv = 1.0f / red[0];

#pragma unroll
  for (int j = 0; j < 8; ++j) {
    const int i = t + j * 256;
    Bt[i * OUT_F + o] =
        rev_w[o * IN_F + i] * (p[j] * inv) * weight[i * OUT_F + o];
  }
}

// ---------------------------------------------------------------------------
// Phase 2: C(BATCH x OUT_F) = A(BATCH x IN_F) * Bt(IN_F x OUT_F), fp32 WMMA.
// 128x128 block tile, BK=32, double-buffered LDS filled by
// GLOBAL_LOAD_ASYNC_TO_LDS_B128 (ASYNCcnt), one barrier per K-tile.
// 8 waves, each owns a 64x32 tile = 4x2 f32 accumulators.
// ---------------------------------------------------------------------------
#define BM 128
#define BN 128
#define BK 32
#define LDT  36   // A LDS row stride (floats): 144B rows, 16B aligned, conflict-free
#define LDBN 136  // B LDS row stride (floats): 544B rows, 16B aligned;
                  // 2*LDBN % 64 == 16 -> half-waves hit disjoint bank ranges

// Issue the async global->LDS copies for one K-tile (8 x b128 per thread).
__device__ __forceinline__ void issue_tile_async(
    const float* __restrict__ A, const float* __restrict__ Bt,
    float* as, float* bs, int m0, int n0, int kb, int t)
{
#pragma unroll
  for (int r = 0; r < 4; ++r) {           // A: 128x32 floats, linear per lane
    const int linear = r * 256 + t;
    const int row = linear >> 3;          // 8 float4 per 32-float row
    const int cg  = (linear & 7) * 4;
    __builtin_amdgcn_global_load_async_to_lds_b128(
        ASG(A + (m0 + row) * IN_F + kb + cg),
        ASL(as + row * LDT + cg), 0, 0);
  }
#pragma unroll
  for (int r = 0; r < 4; ++r) {           // B: 32x128 floats, straight row copy
    const int linear = r * 256 + t;
    const int row = linear >> 5;          // k within tile
    const int cg  = (linear & 31) * 4;    // n within tile
    __builtin_amdgcn_global_load_async_to_lds_b128(
        ASG(Bt + (kb + row) * OUT_F + n0 + cg),
        ASL(bs + row * LDBN + cg), 0, 0);
  }
}

__global__ __launch_bounds__(256) void revmask_gemm(
    const float* __restrict__ A,   // (BATCH, IN_F) row-major
    const float* __restrict__ Bt,  // (IN_F, OUT_F) row-major
    float* __restrict__ C)         // (BATCH, OUT_F) row-major
{
  __shared__ float As[2][BM * LDT];   // [m][k], k contiguous
  __shared__ float Bs[2][BK * LDBN];  // [k][n], n contiguous

  const int t     = threadIdx.x;
  const int lane  = t & 31;
  const int wave  = t >> 5;
  const int waveM = (wave >> 2) * 64;
  const int waveN = (wave & 3) * 32;
  const int m0 = blockIdx.y * BM;
  const int n0 = blockIdx.x * BN;

  const int l15   = lane & 15;
  const int khalf = (lane >> 4) * 2;  // frag: low lanes K0..1, high lanes K2..3

  v8f acc[4][2];
#pragma unroll
  for (int i = 0; i < 4; ++i)
#pragma unroll
    for (int j = 0; j < 2; ++j)
      acc[i][j] = (v8f)0.0f;

  // Prologue: start tile 0 streaming into buffer 0.
  issue_tile_async(A, Bt, As[0], Bs[0], m0, n0, 0, t);

  int buf = 0;
  for (int kb = 0; kb < IN_F; kb += BK) {
    __builtin_amdgcn_s_wait_asynccnt(0);  // this wave's tile-kb copies landed
    __syncthreads();                      // everyone's landed; prev compute done

    if (kb + BK < IN_F)                   // overlap next tile with compute
      issue_tile_async(A, Bt, As[buf ^ 1], Bs[buf ^ 1], m0, n0, kb + BK, t);

    const float* as = As[buf];
    const float* bs = Bs[buf];
#pragma unroll
    for (int kk = 0; kk < BK; kk += 4) {
      v2f af[4], bf[2];
#pragma unroll
      for (int i = 0; i < 4; ++i)
        af[i] = *(const v2f*)&as[(waveM + i * 16 + l15) * LDT + kk + khalf];
#pragma unroll
      for (int j = 0; j < 2; ++j) {
        const float* bp = &bs[(kk + khalf) * LDBN + waveN + j * 16 + l15];
        v2f b; b.x = bp[0]; b.y = bp[LDBN];   // fuses to ds_load_2addr_b32
        bf[j] = b;
      }
#pragma unroll
      for (int i = 0; i < 4; ++i)
#pragma unroll
        for (int j = 0; j < 2; ++j)
          acc[i][j] = __builtin_amdgcn_wmma_f32_16x16x4_f32(
              false, af[i], false, bf[j], (short)0, acc[i][j], false, false);
    }
    buf ^= 1;
  }

  // Epilogue: C/D layout = VGPR r: lanes0-15 M=r, lanes16-31 M=r+8; N=lane&15
  const int mhalf = (lane >> 4) * 8;
#pragma unroll
  for (int i = 0; i < 4; ++i) {
#pragma unroll
    for (int j = 0; j < 2; ++j) {
      const int col = n0 + waveN + j * 16 + l15;
#pragma unroll
      for (int r = 0; r < 8; ++r) {
        const int row = m0 + waveM + i * 16 + mhalf + r;
        C[row * OUT_F + col] = acc[i][j][r];
      }
    }
  }
}

// ---------------------------------------------------------------------------
extern "C" void kernel_launch(void* const* d_in, const int* in_sizes, int n_in,
                              void* d_out, int out_size, void* d_ws, size_t ws_size,
                              hipStream_t stream) {
  const float* inp    = (const float*)d_in[0];  // (8192, 2048)
  const float* weight = (const float*)d_in[1];  // (2048, 2048) = (IN, OUT)
  const float* rev_w  = (const float*)d_in[2];  // (2048, 2048) = (OUT, IN)
  const float* mask_w = (const float*)d_in[3];  // (2048, 2048) = (OUT, IN)
  float* out = (float*)d_out;                   // (8192, 2048)
  float* Bt  = (float*)d_ws;                    // IN_F*OUT_F*4 = 16 MB scratch

  revmask_prep<<<OUT_F, 256, 0, stream>>>(weight, rev_w, mask_w, Bt);

  dim3 grid(OUT_F / BN, BATCH / BM);  // (16, 64)
  revmask_gemm<<<grid, 256, 0, stream>>>(inp, Bt, out);
}